// CustomMSELoss_57260503990383
// MI455X (gfx1250) — compile-verified
//
#include <hip/hip_runtime.h>

typedef __attribute__((ext_vector_type(2))) float v2f;
typedef __attribute__((ext_vector_type(8))) float v8f;

#define NBLOCKS      16384   // 2 * 8192 Hadamard blocks of 64x64
#define WG_SIZE      256
#define WAVES_PER_WG 8       // wave32
#define NUM_WG       2048    // NUM_WG * WAVES_PER_WG == NBLOCKS

// SATD: sum |H64 @ (orig - pred)| over 16384 blocks of 64x64 fp32.
// H64 = H4 (x) H16 (Sylvester), so per 16-column panel:
//   U[c] = H16 @ D[c*16 .. c*16+15, :]   (fp32 WMMA 16x16x4, A = H16 in regs)
//   t[a] = sum_c H4[a][c] * U[c]         (VALU butterfly)
//   acc += sum |t[a]|
// Loads are batched per panel (64 NT loads per lane in flight) before the WMMA chain.
__global__ __launch_bounds__(WG_SIZE)
void satd_hadamard_kernel(const float* __restrict__ orig,
                          const float* __restrict__ pred,
                          float* __restrict__ out) {
    const int lane = threadIdx.x & 31;
    const int wave = threadIdx.x >> 5;
    const int i    = lane & 15;   // tile row (A) / tile col (B,D)
    const int hi   = lane >> 4;   // half-wave select

    // A fragments: H16 K-slices. afrag[f][v] = H16[i][4f + 2hi + v],
    // H16[i][j] = (-1)^popcount(i & j)  (exact +-1 in fp32).
    v2f afrag[4];
#pragma unroll
    for (int f = 0; f < 4; ++f) {
#pragma unroll
        for (int v = 0; v < 2; ++v) {
            int j = 4 * f + 2 * hi + v;
            afrag[f][v] = (__popc(i & j) & 1) ? -1.0f : 1.0f;
        }
    }

    float acc = 0.0f;

    const int gwave = blockIdx.x * WAVES_PER_WG + wave;
    for (int b = gwave; b < NBLOCKS; b += NUM_WG * WAVES_PER_WG) {
        const float* ob = orig + (size_t)b * 4096 + i;  // fold lane column in
        const float* pb = pred + (size_t)b * 4096 + i;
#pragma unroll
        for (int kt = 0; kt < 4; ++kt) {                // 16-column panels
            const float* o = ob + kt * 16;
            const float* p = pb + kt * 16;

            // Phase 1: batch all panel loads (high MLP, non-temporal —
            // single-pass stream, keep it out of the caches).
            float ov[32], pv[32];
#pragma unroll
            for (int t = 0; t < 16; ++t) {              // t = c*4 + f
                const int j0 = 4 * t + 2 * hi;          // = c*16 + f*4 + 2hi
                ov[2 * t]     = __builtin_nontemporal_load(o + (size_t)j0 * 64);
                ov[2 * t + 1] = __builtin_nontemporal_load(o + (size_t)(j0 + 1) * 64);
                pv[2 * t]     = __builtin_nontemporal_load(p + (size_t)j0 * 64);
                pv[2 * t + 1] = __builtin_nontemporal_load(p + (size_t)(j0 + 1) * 64);
            }

            // Phase 2: diff fragments.
            v2f bf[16];
#pragma unroll
            for (int t = 0; t < 16; ++t) {
                bf[t][0] = ov[2 * t]     - pv[2 * t];
                bf[t][1] = ov[2 * t + 1] - pv[2 * t + 1];
            }

            // Phase 3: WMMA chains (f32 in, f32 accumulate: exact math).
            v8f U[4];
#pragma unroll
            for (int c = 0; c < 4; ++c) {
                U[c] = v8f{};
#pragma unroll
                for (int f = 0; f < 4; ++f) {
                    U[c] = __builtin_amdgcn_wmma_f32_16x16x4_f32(
                        /*neg_a=*/false, afrag[f],
                        /*neg_b=*/false, bf[c * 4 + f],
                        /*c_mod=*/(short)0, U[c],
                        /*reuse_a=*/false, /*reuse_b=*/false);
                }
            }

            // Phase 4: H4 butterfly across row-groups + |.| accumulate.
#pragma unroll
            for (int s = 0; s < 8; ++s) {
                float pq = U[0][s] + U[1][s];
                float mq = U[0][s] - U[1][s];
                float qq = U[2][s] + U[3][s];
                float rq = U[2][s] - U[3][s];
                acc += fabsf(pq + qq) + fabsf(pq - qq)
                     + fabsf(mq + rq) + fabsf(mq - rq);
            }
        }
    }

    // Wave32 reduction.
#pragma unroll
    for (int m = 16; m >= 1; m >>= 1)
        acc += __shfl_xor(acc, m, 32);

    __shared__ float sred[WAVES_PER_WG];
    if (lane == 0) sred[wave] = acc;
    __syncthreads();
    if (threadIdx.x == 0) {
        float t = 0.0f;
#pragma unroll
        for (int w = 0; w < WAVES_PER_WG; ++w) t += sred[w];
        atomicAdd(out, t);
    }
}

extern "C" void kernel_launch(void* const* d_in, const int* in_sizes, int n_in,
                              void* d_out, int out_size, void* d_ws, size_t ws_size,
                              hipStream_t stream) {
    const float* orig = (const float*)d_in[0];
    const float* pred = (const float*)d_in[1];
    float* out = (float*)d_out;

    // Harness poisons d_out; zero the scalar accumulator (capture-safe).
    hipMemsetAsync(out, 0, sizeof(float), stream);

    satd_hadamard_kernel<<<NUM_WG, WG_SIZE, 0, stream>>>(orig, pred, out);
}